// TwoBranch_90117003805454
// MI455X (gfx1250) — compile-verified
//
#include <hip/hip_runtime.h>
#include <math.h>

#define B_      4
#define C_      64
#define H_      256
#define W_      256
#define HW_     (H_ * W_)
#define HEADS_  8
#define NCHUNK  8

typedef __attribute__((ext_vector_type(2))) float v2f;
typedef __attribute__((ext_vector_type(8))) float v8f;

static __device__ __forceinline__ v8f wmma4(v2f a, v2f b, v8f c) {
  // D(16x16 f32) = A(16x4 f32) * B(4x16 f32) + C   -> v_wmma_f32_16x16x4_f32
  return __builtin_amdgcn_wmma_f32_16x16x4_f32(false, a, false, b, (short)0, c,
                                               false, false);
}

// CDNA5 async copy: memory -> LDS without VGPR round trip (ASYNCcnt tracked).
static __device__ __forceinline__ void async_ld32(unsigned lds_off,
                                                  const float* g) {
  asm volatile("global_load_async_to_lds_b32 %0, %1, off" ::"v"(lds_off),
               "v"((unsigned long long)(uintptr_t)g)
               : "memory");
}
static __device__ __forceinline__ void wait_async0() {
  asm volatile("s_wait_asynccnt 0x0" ::: "memory");
}

// -------- weight repack for 3x3 implicit GEMM: W[co][ci][f] -> WP[co][f][ci] --
__global__ __launch_bounds__(256) void pack_w3(const float* __restrict__ W,
                                               float* __restrict__ WP,
                                               int Cout, int Cin) {
  int K = Cin * 9;
  int i = blockIdx.x * 256 + threadIdx.x;
  if (i >= Cout * K) return;
  int co = i / K, rem = i % K;
  int f = rem / Cin, ci = rem % Cin;
  WP[i] = W[(size_t)co * K + ci * 9 + f];
}

// -------- 3x3 conv, pad=1, implicit GEMM via WMMA f32 16x16x4 ----------------
// inB supplies channels [Ca, Cin) (for the concat conv); may be nullptr.
__global__ __launch_bounds__(256) void conv3x3_wmma(
    const float* __restrict__ inA, const float* __restrict__ inB, int Ca,
    int Cin, const float* __restrict__ WP, const float* __restrict__ bias,
    float* __restrict__ out, int Cout) {
  __shared__ float patch[3 * 34 * 17];  // [row][col][ci16], pad 17 vs bank conflicts

  int tx = (blockIdx.x & 7) * 32;   // W_/32 == 8 tiles per row
  int ty = blockIdx.x >> 3;
  int n  = blockIdx.y;
  int coG = blockIdx.z * 64;

  int wv = threadIdx.x >> 5, lane = threadIdx.x & 31;
  int lo = lane & 15, hi = lane >> 4;
  int coTile = coG + (wv >> 1) * 16;
  int pixSub = (wv & 1) * 16;
  int K = Cin * 9;
  int Cb = Cin - Ca;

  v8f acc = {};
  for (int cc = 0; cc < Cin; cc += 16) {
    const float* wrow = WP + (size_t)(coTile + lo) * K + cc;

    // ---- A-panel register preload for this chunk (independent of LDS; its
    //      global latency overlaps the async staging + barrier below) ----
    v2f areg[36];
#pragma unroll
    for (int f = 0; f < 9; f++)
#pragma unroll
      for (int s = 0; s < 4; s++)
        areg[f * 4 + s] = *(const v2f*)(wrow + f * Cin + s * 4 + 2 * hi);

    // ---- stage 16-channel input halo patch into LDS via async copy ----
    for (int e = threadIdx.x; e < 3 * 34 * 16; e += 256) {
      int ci = e & 15;
      int rc = e >> 4;
      int c = rc % 34, r = rc / 34;
      int gy = ty + r - 1, gx = tx + c - 1;
      unsigned laddr = (unsigned)(uintptr_t)&patch[(r * 34 + c) * 17 + ci];
      if ((unsigned)gy < (unsigned)H_ && (unsigned)gx < (unsigned)W_) {
        int cig = cc + ci;
        const float* src =
            (cig < Ca) ? &inA[(((size_t)n * Ca + cig) * H_ + gy) * W_ + gx]
                       : &inB[(((size_t)n * Cb + (cig - Ca)) * H_ + gy) * W_ + gx];
        async_ld32(laddr, src);
      } else {
        patch[(r * 34 + c) * 17 + ci] = 0.f;  // halo zero-fill
      }
    }
    wait_async0();
    __syncthreads();

    // ---- compute: per tap-group load 12 B pairs, then 12 back-to-back wmma --
#pragma unroll
    for (int fg = 0; fg < 3; fg++) {
      v2f breg[12];
#pragma unroll
      for (int j = 0; j < 12; j++) {
        int f = fg * 3 + j / 4, s = j % 4;
        int ky = f / 3, kx = f % 3;
        int cio = s * 4 + 2 * hi;
        int bbase = (ky * 34 + pixSub + lo + kx) * 17 + cio;
        breg[j].x = patch[bbase];
        breg[j].y = patch[bbase + 1];
      }
#pragma unroll
      for (int j = 0; j < 12; j++) acc = wmma4(areg[fg * 12 + j], breg[j], acc);
    }
    __syncthreads();
  }

  int px = tx + pixSub + lo;
#pragma unroll
  for (int r = 0; r < 8; r++) {
    int co = coTile + r + 8 * hi;   // D layout: vgpr r -> M=r (lanes0-15), r+8 (lanes16-31)
    float v = acc[r] + (bias ? bias[co] : 0.f);
    out[(((size_t)n * Cout + co) * H_ + ty) * W_ + px] = v;
  }
}

// -------- 1x1 conv (GEMM over channels, Cin=64) via WMMA, optional residual --
__global__ __launch_bounds__(256) void conv1x1_wmma(
    const float* __restrict__ in, const float* __restrict__ Wk,
    const float* __restrict__ res, float* __restrict__ out, int Cout) {
  int n = blockIdx.y;
  int coG = blockIdx.z * 64;
  int wv = threadIdx.x >> 5, lane = threadIdx.x & 31;
  int lo = lane & 15, hi = lane >> 4;
  int coTile = coG + (wv >> 1) * 16;
  int pix = blockIdx.x * 32 + (wv & 1) * 16 + lo;

  const float* ib = in + (size_t)n * C_ * HW_ + pix;
  const float* wrow = Wk + (size_t)(coTile + lo) * C_;

  // full K panel preload (16 steps), then 16 uninterrupted wmma
  v2f areg[16], breg[16];
#pragma unroll
  for (int t = 0; t < 16; t++) {
    int k1 = t * 4 + 2 * hi;
    areg[t] = *(const v2f*)(wrow + k1);
    breg[t].x = ib[(size_t)k1 * HW_];
    breg[t].y = ib[(size_t)(k1 + 1) * HW_];
  }
  v8f acc = {};
#pragma unroll
  for (int t = 0; t < 16; t++) acc = wmma4(areg[t], breg[t], acc);

#pragma unroll
  for (int r = 0; r < 8; r++) {
    int co = coTile + r + 8 * hi;
    size_t oi = ((size_t)n * Cout + co) * HW_ + pix;
    float v = acc[r];
    if (res) v += res[oi];
    out[oi] = v;
  }
}

// -------- depthwise 3x3, pad=1, no bias (memory-bound stencil) ---------------
__global__ __launch_bounds__(256) void dwconv3x3(const float* __restrict__ in,
                                                 const float* __restrict__ w,
                                                 float* __restrict__ out,
                                                 int Cch) {
  size_t i = (size_t)blockIdx.x * 256 + threadIdx.x;
  int x = i & (W_ - 1);
  int y = (i >> 8) & (H_ - 1);
  size_t bc = i >> 16;
  int c = (int)(bc % Cch);
  const float* wp = w + (size_t)c * 9;
  const float* ip = in + bc * HW_;
  float s = 0.f;
#pragma unroll
  for (int ky = 0; ky < 3; ky++)
#pragma unroll
    for (int kx = 0; kx < 3; kx++) {
      int gy = y + ky - 1, gx = x + kx - 1;
      if ((unsigned)gy < (unsigned)H_ && (unsigned)gx < (unsigned)W_)
        s += wp[ky * 3 + kx] * ip[gy * W_ + gx];
    }
  out[i] = s;
}

// -------- Gram partials: S[8][8], |q|^2, |k|^2 per (bh, chunk) ---------------
__global__ __launch_bounds__(256) void gram_partial(const float* __restrict__ q,
                                                    const float* __restrict__ kk,
                                                    float* __restrict__ SP) {
  int chunk = blockIdx.x, bh = blockIdx.y;
  int b = bh >> 3, h = bh & 7;
  const float* qb = q + ((size_t)b * C_ + h * 8) * HW_;
  const float* kb = kk + ((size_t)b * (2 * C_) + h * 8) * HW_;  // k = first C of kv

  float S[8][8], nq[8], nk[8];
#pragma unroll
  for (int c = 0; c < 8; c++) {
    nq[c] = 0.f; nk[c] = 0.f;
#pragma unroll
    for (int d = 0; d < 8; d++) S[c][d] = 0.f;
  }

  int pend = (chunk + 1) * (HW_ / NCHUNK);
  for (int p = chunk * (HW_ / NCHUNK) + threadIdx.x; p < pend; p += 256) {
    float qv[8], kv[8];
#pragma unroll
    for (int c = 0; c < 8; c++) qv[c] = qb[(size_t)c * HW_ + p];
#pragma unroll
    for (int d = 0; d < 8; d++) kv[d] = kb[(size_t)d * HW_ + p];
#pragma unroll
    for (int c = 0; c < 8; c++) {
      nq[c] += qv[c] * qv[c];
#pragma unroll
      for (int d = 0; d < 8; d++) S[c][d] += qv[c] * kv[d];
    }
#pragma unroll
    for (int d = 0; d < 8; d++) nk[d] += kv[d] * kv[d];
  }

  // wave32 reduction
#pragma unroll
  for (int c = 0; c < 8; c++) {
#pragma unroll
    for (int d = 0; d < 8; d++)
      for (int off = 16; off > 0; off >>= 1) S[c][d] += __shfl_xor(S[c][d], off);
    for (int off = 16; off > 0; off >>= 1) nq[c] += __shfl_xor(nq[c], off);
    for (int off = 16; off > 0; off >>= 1) nk[c] += __shfl_xor(nk[c], off);
  }

  __shared__ float red[8][80];
  int wv = threadIdx.x >> 5, lane = threadIdx.x & 31;
  if (lane == 0) {
#pragma unroll
    for (int c = 0; c < 8; c++) {
#pragma unroll
      for (int d = 0; d < 8; d++) red[wv][c * 8 + d] = S[c][d];
      red[wv][64 + c] = nq[c];
      red[wv][72 + c] = nk[c];
    }
  }
  __syncthreads();
  if (threadIdx.x < 80) {
    float s = 0.f;
#pragma unroll
    for (int w = 0; w < 8; w++) s += red[w][threadIdx.x];
    SP[((size_t)bh * NCHUNK + chunk) * 80 + threadIdx.x] = s;
  }
}

// -------- combine partials (fixed order -> deterministic), l2norm, softmax ---
__global__ __launch_bounds__(128) void attn_finalize(const float* __restrict__ SP,
                                                     const float* __restrict__ temp,
                                                     float* __restrict__ ATT) {
  int bh = blockIdx.x, h = bh & 7;
  int tid = threadIdx.x;
  __shared__ float tot[80];
  __shared__ float logit[64];
  if (tid < 80) {
    float s = 0.f;
    for (int c = 0; c < NCHUNK; c++) s += SP[((size_t)bh * NCHUNK + c) * 80 + tid];
    tot[tid] = s;
  }
  __syncthreads();
  if (tid < 64) {
    int c = tid >> 3, d = tid & 7;
    float nq = fmaxf(sqrtf(tot[64 + c]), 1e-12f);
    float nk = fmaxf(sqrtf(tot[72 + d]), 1e-12f);
    logit[tid] = tot[tid] / (nq * nk) * temp[h];
  }
  __syncthreads();
  if (tid < 8) {
    float m = -1e30f;
    for (int d = 0; d < 8; d++) m = fmaxf(m, logit[tid * 8 + d]);
    float e[8], sum = 0.f;
    for (int d = 0; d < 8; d++) { e[d] = expf(logit[tid * 8 + d] - m); sum += e[d]; }
    for (int d = 0; d < 8; d++) ATT[(size_t)bh * 64 + tid * 8 + d] = e[d] / sum;
  }
}

// -------- out[c,p] = sum_d attn[c][d] * v[d,p]  (bandwidth bound) ------------
__global__ __launch_bounds__(256) void attn_apply(const float* __restrict__ ATT,
                                                  const float* __restrict__ kvd,
                                                  float* __restrict__ o) {
  int bh = blockIdx.y, b = bh >> 3, h = bh & 7;
  __shared__ float a[64];
  if (threadIdx.x < 64) a[threadIdx.x] = ATT[(size_t)bh * 64 + threadIdx.x];
  __syncthreads();
  const float* vb = kvd + ((size_t)b * (2 * C_) + C_ + h * 8) * HW_;  // v = 2nd half
  float* ob = o + ((size_t)b * C_ + h * 8) * HW_;
#pragma unroll
  for (int it = 0; it < 4; it++) {
    int p = blockIdx.x * 1024 + it * 256 + threadIdx.x;
    float v[8];
#pragma unroll
    for (int d = 0; d < 8; d++) v[d] = vb[(size_t)d * HW_ + p];
#pragma unroll
    for (int c = 0; c < 8; c++) {
      float s = 0.f;
#pragma unroll
      for (int d = 0; d < 8; d++) s += a[c * 8 + d] * v[d];
      ob[(size_t)c * HW_ + p] = s;
    }
  }
}

// -------- sigmoid gates + blend + nan_to_num ---------------------------------
__global__ __launch_bounds__(256) void final_fuse(const float* __restrict__ f2,
                                                  const float* __restrict__ s2,
                                                  const float* __restrict__ t2,
                                                  float* __restrict__ out) {
  size_t i = (size_t)blockIdx.x * 256 + threadIdx.x;
  int p = (int)(i & (size_t)(HW_ - 1));
  size_t bc = i >> 16;
  int b = (int)(bc >> 6), c = (int)(bc & 63);
  float g1 = t2[(((size_t)b * (2 * C_)) + c) * HW_ + p];
  float g2 = t2[(((size_t)b * (2 * C_)) + C_ + c) * HW_ + p];
  float a1 = 1.f / (1.f + expf(-g1));
  float a2 = 1.f / (1.f + expf(-g2));
  float r = f2[i] * a1 + s2[i] * a2;
  out[i] = isfinite(r) ? r : 1e-5f;
}

extern "C" void kernel_launch(void* const* d_in, const int* in_sizes, int n_in,
                              void* d_out, int out_size, void* d_ws, size_t ws_size,
                              hipStream_t stream) {
  (void)in_sizes; (void)n_in; (void)out_size; (void)ws_size;
  const float* spa0  = (const float*)d_in[0];
  const float* fre0  = (const float*)d_in[1];
  const float* W_fre = (const float*)d_in[2];
  const float* b_fre = (const float*)d_in[3];
  const float* W_spa = (const float*)d_in[4];
  const float* b_spa = (const float*)d_in[5];
  const float* temp  = (const float*)d_in[6];
  const float* Wq    = (const float*)d_in[7];
  const float* Wq_dw = (const float*)d_in[8];
  const float* Wkv   = (const float*)d_in[9];
  const float* Wkv_dw= (const float*)d_in[10];
  const float* Wproj = (const float*)d_in[11];
  const float* Wf1   = (const float*)d_in[12];
  const float* bf1   = (const float*)d_in[13];
  const float* Wf2   = (const float*)d_in[14];
  const float* bf2   = (const float*)d_in[15];
  float* out = (float*)d_out;
  float* ws  = (float*)d_ws;

  const size_t U = (size_t)B_ * C_ * HW_;     // one BCHW fp32 tensor
  float* fre1 = ws;
  float* spa1 = ws + U;
  float* Qb   = ws + 2 * U;
  float* QD   = ws + 3 * U;
  float* KV   = ws + 4 * U;                   // 2U
  float* KVD  = ws + 6 * U;                   // 2U
  float* F2   = ws + 8 * U;
  float* S2   = ws + 9 * U;
  float* WP   = ws + 10 * U;                  // packed weights (<=147456 floats)
  float* SP   = WP + 147456;                  // 32*NCHUNK*80 partials
  float* ATT  = SP + (size_t)32 * NCHUNK * 80;
  float* T1   = fre1;                         // fre1 dead after attention pass 1
  float* T2   = KV;                           // KV dead after attention pass 2
  float* O    = Qb;                           // Qb dead after its depthwise conv

  dim3 blk(256);

  auto c3 = [&](const float* inA, const float* inB, int Ca, int Cin,
                const float* Wraw, const float* bias, float* o_, int Cout) {
    int total = Cout * Cin * 9;
    pack_w3<<<dim3((total + 255) / 256), blk, 0, stream>>>(Wraw, WP, Cout, Cin);
    conv3x3_wmma<<<dim3((W_ / 32) * H_, B_, Cout / 64), blk, 0, stream>>>(
        inA, inB, Ca, Cin, WP, bias, o_, Cout);
  };
  auto c1 = [&](const float* in, const float* Wk, const float* res,
                float* o_, int Cout) {
    conv1x1_wmma<<<dim3(HW_ / 32, B_, Cout / 64), blk, 0, stream>>>(
        in, Wk, res, o_, Cout);
  };
  auto dw = [&](const float* in, const float* w, float* o_, int Cch) {
    size_t tot = (size_t)B_ * Cch * HW_;
    dwconv3x3<<<dim3((unsigned)(tot / 256)), blk, 0, stream>>>(in, w, o_, Cch);
  };
  auto attn = [&](const float* x, const float* y, const float* res, float* dst) {
    c1(y, Wkv, nullptr, KV, 2 * C_);
    dw(KV, Wkv_dw, KVD, 2 * C_);
    c1(x, Wq, nullptr, Qb, C_);
    dw(Qb, Wq_dw, QD, C_);
    gram_partial<<<dim3(NCHUNK, B_ * HEADS_), blk, 0, stream>>>(QD, KVD, SP);
    attn_finalize<<<dim3(B_ * HEADS_), dim3(128), 0, stream>>>(SP, temp, ATT);
    attn_apply<<<dim3(HW_ / 1024, B_ * HEADS_), blk, 0, stream>>>(ATT, KVD, O);
    c1(O, Wproj, res, dst, C_);
  };

  // fre = conv3x3(fre); spa = conv3x3(spa)
  c3(fre0, nullptr, C_, C_, W_fre, b_fre, fre1, C_);
  c3(spa0, nullptr, C_, C_, W_spa, b_spa, spa1, C_);
  // fre = att(fre, spa) + fre ; spa = att(spa, fre) + spa
  attn(fre1, spa1, fre1, F2);
  attn(spa1, F2, spa1, S2);
  // fuse = sigmoid(conv3x3(conv3x3(cat(fre, spa))))
  c3(F2, S2, C_, 2 * C_, Wf1, bf1, T1, C_);
  c3(T1, nullptr, C_, C_, Wf2, bf2, T2, 2 * C_);
  final_fuse<<<dim3((unsigned)(U / 256)), blk, 0, stream>>>(F2, S2, T2, out);
}